// EdgeGATConv_31293131718793
// MI455X (gfx1250) — compile-verified
//
#include <hip/hip_runtime.h>

#define GAT_HEADS 4
#define GAT_OUTC 16
#define GAT_HC 64
#define GAT_INC 128
#define GAT_EDGED 16
#define GAT_SLOPE 0.2f

typedef __attribute__((ext_vector_type(2))) float v2f;
typedef __attribute__((ext_vector_type(8))) float v8f;

// ---------------------------------------------------------------------------
// float atomic max via sign-split integer atomics (init value must be -inf)
// ---------------------------------------------------------------------------
__device__ __forceinline__ void atomicMaxF32(float* addr, float val) {
    if (val >= 0.0f) {
        atomicMax((int*)addr, __float_as_int(val));
    } else {
        atomicMin((unsigned int*)addr, (unsigned int)__float_as_int(val));
    }
}

// ---------------------------------------------------------------------------
// Init: zero out [N*64], zero s [N*4], m = -inf [N*4]
// ---------------------------------------------------------------------------
__global__ void egat_init(float* __restrict__ out, float* __restrict__ m,
                          float* __restrict__ s, int nout, int nh) {
    int i = blockIdx.x * blockDim.x + threadIdx.x;
    if (i < nout) out[i] = 0.0f;
    if (i < nh) { s[i] = 0.0f; m[i] = -__builtin_inff(); }
}

// ---------------------------------------------------------------------------
// Fold attention vectors into the linear weights:
//   w_as[d][h] = sum_c W_src[d][h*16+c]*att_src[h][c]   (128x4)
//   w_ad[d][h] = sum_c W_dst[d][h*16+c]*att_dst[h][c]   (128x4)
//   w_ae[d][h] = sum_c W_edge[d][h*16+c]*att_edge[h][c] (16x4)
//   csts[0..3]=b_src.att, [4..7]=b_dst.att, [8..11]=b_edge.att
// ---------------------------------------------------------------------------
__global__ void egat_fold(const float* __restrict__ W_src, const float* __restrict__ b_src,
                          const float* __restrict__ W_dst, const float* __restrict__ b_dst,
                          const float* __restrict__ W_edge, const float* __restrict__ b_edge,
                          const float* __restrict__ att_src, const float* __restrict__ att_dst,
                          const float* __restrict__ att_edge,
                          float* __restrict__ w_as, float* __restrict__ w_ad,
                          float* __restrict__ w_ae, float* __restrict__ csts) {
    int d = threadIdx.x;
    #pragma unroll
    for (int h = 0; h < GAT_HEADS; h++) {
        float ss = 0.0f, sd = 0.0f;
        #pragma unroll
        for (int c = 0; c < GAT_OUTC; c++) {
            ss += W_src[d * GAT_HC + h * GAT_OUTC + c] * att_src[h * GAT_OUTC + c];
            sd += W_dst[d * GAT_HC + h * GAT_OUTC + c] * att_dst[h * GAT_OUTC + c];
        }
        w_as[d * GAT_HEADS + h] = ss;
        w_ad[d * GAT_HEADS + h] = sd;
        if (d < GAT_EDGED) {
            float se = 0.0f;
            #pragma unroll
            for (int c = 0; c < GAT_OUTC; c++)
                se += W_edge[d * GAT_HC + h * GAT_OUTC + c] * att_edge[h * GAT_OUTC + c];
            w_ae[d * GAT_HEADS + h] = se;
        }
        if (d == 0) {
            float cs = 0.0f, cd = 0.0f, ce = 0.0f;
            #pragma unroll
            for (int c = 0; c < GAT_OUTC; c++) {
                cs += b_src[h * GAT_OUTC + c] * att_src[h * GAT_OUTC + c];
                cd += b_dst[h * GAT_OUTC + c] * att_dst[h * GAT_OUTC + c];
                ce += b_edge[h * GAT_OUTC + c] * att_edge[h * GAT_OUTC + c];
            }
            csts[h] = cs; csts[4 + h] = cd; csts[8 + h] = ce;
        }
    }
}

// ---------------------------------------------------------------------------
// src_feat = x @ W_src + b_src using V_WMMA_F32_16X16X4_F32 (wave32).
// Block = 128 threads (4 waves). Each wave computes a 16x64 tile.
//
// W_src is staged into LDS in K-pair-interleaved layout:
//   sW[(k>>1)*128 + col*2 + (k&1)] = W[k*64 + col]
// so each B operand (K=k+koff, K=k+koff+1 for a fixed col; k+koff is always
// even) is one contiguous 8-byte-aligned ds_load_b64 landing directly in an
// even-aligned VGPR pair for the WMMA — no repacking movs.
//
// A 16x4 f32 layout: lane%16 = M; lanes<16 hold K={0,1}, lanes>=16 K={2,3}.
// B  4x16 f32 layout: lane%16 = N; VGPR0: K=0/2, VGPR1: K=1/3 (lo/hi half).
// C 16x16 f32 layout: VGPR v -> M = v + (lane<16?0:8), N = lane%16.
// ---------------------------------------------------------------------------
__global__ __launch_bounds__(128) void egat_gemm_src_wmma(
        const float* __restrict__ x, const float* __restrict__ W,
        const float* __restrict__ bias, float* __restrict__ out, int n) {
    __shared__ float sW[GAT_INC * GAT_HC];  // 32 KB, pair-interleaved

    int t = threadIdx.x;
    for (int idx = t; idx < GAT_INC * GAT_HC; idx += 128) {
        int k = idx >> 6;          // 0..127
        int col = idx & 63;        // 0..63
        sW[(k >> 1) * 128 + col * 2 + (k & 1)] = W[idx];
    }
    __syncthreads();

    const int wave = t >> 5;
    const int lane = t & 31;
    const int col  = lane & 15;
    const int koff = (lane < 16) ? 0 : 2;

    const int rowBase = blockIdx.x * 64 + wave * 16;
    int row = rowBase + col;                 // A-matrix row for this lane
    int rowc = row < n ? row : (n - 1);      // clamp (stores are guarded)

    v8f c0 = {}, c1 = {}, c2 = {}, c3 = {};
    const float* xrow = x + (size_t)rowc * GAT_INC;

    #pragma unroll 4
    for (int k = 0; k < GAT_INC; k += 4) {
        float2 av = *(const float2*)(xrow + k + koff);
        v2f a; a.x = av.x; a.y = av.y;
        // base of the K-pair row in pair-interleaved LDS
        const float* bp = sW + ((k + koff) >> 1) * 128 + col * 2;
        v2f b0 = *(const v2f*)(bp +  0);   // cols  0..15
        v2f b1 = *(const v2f*)(bp + 32);   // cols 16..31
        v2f b2 = *(const v2f*)(bp + 64);   // cols 32..47
        v2f b3 = *(const v2f*)(bp + 96);   // cols 48..63
        c0 = __builtin_amdgcn_wmma_f32_16x16x4_f32(false, a, false, b0, (short)0, c0, false, false);
        c1 = __builtin_amdgcn_wmma_f32_16x16x4_f32(false, a, false, b1, (short)0, c1, false, false);
        c2 = __builtin_amdgcn_wmma_f32_16x16x4_f32(false, a, false, b2, (short)0, c2, false, false);
        c3 = __builtin_amdgcn_wmma_f32_16x16x4_f32(false, a, false, b3, (short)0, c3, false, false);
    }

    const float bb0 = bias[ 0 + col];
    const float bb1 = bias[16 + col];
    const float bb2 = bias[32 + col];
    const float bb3 = bias[48 + col];
    const int mbase = rowBase + ((lane < 16) ? 0 : 8);
    #pragma unroll
    for (int v = 0; v < 8; v++) {
        int r = mbase + v;
        if (r < n) {
            float* orow = out + (size_t)r * GAT_HC;
            orow[ 0 + col] = c0[v] + bb0;
            orow[16 + col] = c1[v] + bb1;
            orow[32 + col] = c2[v] + bb2;
            orow[48 + col] = c3[v] + bb3;
        }
    }
}

// ---------------------------------------------------------------------------
// Per-node attention logits: alpha_src[n][h], alpha_dst[n][h] = x[n] @ fold + c
// one wave (32 lanes) per node; lane handles 4 contiguous input dims.
// ---------------------------------------------------------------------------
__global__ __launch_bounds__(256) void egat_node_alpha(
        const float* __restrict__ x, const float* __restrict__ w_as,
        const float* __restrict__ w_ad, const float* __restrict__ csts,
        float* __restrict__ alpha_src, float* __restrict__ alpha_dst, int n) {
    int node = (blockIdx.x * blockDim.x + threadIdx.x) >> 5;
    int lane = threadIdx.x & 31;
    if (node >= n) return;

    float4 xv = *(const float4*)(x + (size_t)node * GAT_INC + lane * 4);
    float xs[4] = {xv.x, xv.y, xv.z, xv.w};

    float as[GAT_HEADS] = {0, 0, 0, 0};
    float ad[GAT_HEADS] = {0, 0, 0, 0};
    #pragma unroll
    for (int j = 0; j < 4; j++) {
        int d = lane * 4 + j;
        #pragma unroll
        for (int h = 0; h < GAT_HEADS; h++) {
            as[h] += xs[j] * w_as[d * GAT_HEADS + h];
            ad[h] += xs[j] * w_ad[d * GAT_HEADS + h];
        }
    }
    #pragma unroll
    for (int off = 16; off > 0; off >>= 1) {
        #pragma unroll
        for (int h = 0; h < GAT_HEADS; h++) {
            as[h] += __shfl_xor(as[h], off, 32);
            ad[h] += __shfl_xor(ad[h], off, 32);
        }
    }
    if (lane == 0) {
        #pragma unroll
        for (int h = 0; h < GAT_HEADS; h++) {
            alpha_src[(size_t)node * GAT_HEADS + h] = as[h] + csts[h];
            alpha_dst[(size_t)node * GAT_HEADS + h] = ad[h] + csts[4 + h];
        }
    }
}

// ---------------------------------------------------------------------------
// Pass 1: alpha[e][h] = leakyrelu(a_src[src] + a_dst[dst] + edge_attr@w_ae + c)
// and atomic float max into m[dst][h].
// ---------------------------------------------------------------------------
__global__ __launch_bounds__(256) void egat_edge_alpha(
        const int* __restrict__ ei, const float* __restrict__ edge_attr,
        const float* __restrict__ alpha_src, const float* __restrict__ alpha_dst,
        const float* __restrict__ w_ae, const float* __restrict__ csts,
        float* __restrict__ alpha_e, float* __restrict__ m, int E) {
    int e = blockIdx.x * blockDim.x + threadIdx.x;
    if (e >= E) return;
    int s = ei[e];
    int d = ei[E + e];

    const float* er = edge_attr + (size_t)e * GAT_EDGED;
    float ea[GAT_EDGED];
    float4 e0 = *(const float4*)(er + 0);
    float4 e1 = *(const float4*)(er + 4);
    float4 e2 = *(const float4*)(er + 8);
    float4 e3 = *(const float4*)(er + 12);
    ea[0]=e0.x; ea[1]=e0.y; ea[2]=e0.z; ea[3]=e0.w;
    ea[4]=e1.x; ea[5]=e1.y; ea[6]=e1.z; ea[7]=e1.w;
    ea[8]=e2.x; ea[9]=e2.y; ea[10]=e2.z; ea[11]=e2.w;
    ea[12]=e3.x; ea[13]=e3.y; ea[14]=e3.z; ea[15]=e3.w;

    #pragma unroll
    for (int h = 0; h < GAT_HEADS; h++) {
        float a = alpha_src[(size_t)s * GAT_HEADS + h] +
                  alpha_dst[(size_t)d * GAT_HEADS + h] + csts[8 + h];
        #pragma unroll
        for (int k = 0; k < GAT_EDGED; k++)
            a += ea[k] * w_ae[k * GAT_HEADS + h];
        a = (a >= 0.0f) ? a : GAT_SLOPE * a;
        alpha_e[(size_t)e * GAT_HEADS + h] = a;
        atomicMaxF32(&m[(size_t)d * GAT_HEADS + h], a);
    }
}

// ---------------------------------------------------------------------------
// Pass 2: a = exp(alpha - m[dst]); store back; atomicAdd into s[dst][h].
// ---------------------------------------------------------------------------
__global__ __launch_bounds__(256) void egat_edge_exp(
        const int* __restrict__ ei, const float* __restrict__ m,
        float* __restrict__ alpha_e, float* __restrict__ ssum, int E) {
    int e = blockIdx.x * blockDim.x + threadIdx.x;
    if (e >= E) return;
    int d = ei[E + e];
    #pragma unroll
    for (int h = 0; h < GAT_HEADS; h++) {
        float a = __expf(alpha_e[(size_t)e * GAT_HEADS + h] -
                         m[(size_t)d * GAT_HEADS + h]);
        alpha_e[(size_t)e * GAT_HEADS + h] = a;
        atomicAdd(&ssum[(size_t)d * GAT_HEADS + h], a);
    }
}

// ---------------------------------------------------------------------------
// Pass 3: out[dst][h*16+c] += src_feat[src][h*16+c] * a/(s+eps)
// 64 lanes per edge (coalesced gather of src row + coalesced fp32 atomics).
// ---------------------------------------------------------------------------
__global__ __launch_bounds__(256) void egat_scatter(
        const int* __restrict__ ei, const float* __restrict__ src_feat,
        const float* __restrict__ alpha_e, const float* __restrict__ ssum,
        float* __restrict__ out, int E) {
    long long tid = (long long)blockIdx.x * blockDim.x + threadIdx.x;
    int e = (int)(tid >> 6);
    int c = (int)(tid & 63);
    if (e >= E) return;
    int s = ei[e];
    int d = ei[E + e];
    int h = c >> 4;
    float w = alpha_e[(size_t)e * GAT_HEADS + h] /
              (ssum[(size_t)d * GAT_HEADS + h] + 1e-16f);
    atomicAdd(&out[(size_t)d * GAT_HC + c],
              src_feat[(size_t)s * GAT_HC + c] * w);
}

// ---------------------------------------------------------------------------
extern "C" void kernel_launch(void* const* d_in, const int* in_sizes, int n_in,
                              void* d_out, int out_size, void* d_ws, size_t ws_size,
                              hipStream_t stream) {
    const float* x         = (const float*)d_in[0];
    const int*   ei        = (const int*)d_in[1];
    const float* edge_attr = (const float*)d_in[2];
    const float* W_src     = (const float*)d_in[3];
    const float* b_src     = (const float*)d_in[4];
    const float* W_dst     = (const float*)d_in[5];
    const float* b_dst     = (const float*)d_in[6];
    const float* W_edge    = (const float*)d_in[7];
    const float* b_edge    = (const float*)d_in[8];
    const float* att_src   = (const float*)d_in[9];
    const float* att_dst   = (const float*)d_in[10];
    const float* att_edge  = (const float*)d_in[11];
    float* out = (float*)d_out;

    const int N = in_sizes[0] / GAT_INC;
    const int E = in_sizes[1] / 2;

    // workspace layout (floats)
    float* ws = (float*)d_ws;
    size_t o = 0;
    float* src_feat  = ws + o; o += (size_t)N * GAT_HC;
    float* alpha_src = ws + o; o += (size_t)N * GAT_HEADS;
    float* alpha_dst = ws + o; o += (size_t)N * GAT_HEADS;
    float* m_max     = ws + o; o += (size_t)N * GAT_HEADS;
    float* s_sum     = ws + o; o += (size_t)N * GAT_HEADS;
    float* alpha_e   = ws + o; o += (size_t)E * GAT_HEADS;
    float* w_as      = ws + o; o += GAT_INC * GAT_HEADS;
    float* w_ad      = ws + o; o += GAT_INC * GAT_HEADS;
    float* w_ae      = ws + o; o += GAT_EDGED * GAT_HEADS;
    float* csts      = ws + o; o += 16;

    // 1) init accumulators
    {
        int nout = N * GAT_HC, nh = N * GAT_HEADS;
        int blocks = (nout + 255) / 256;
        egat_init<<<blocks, 256, 0, stream>>>(out, m_max, s_sum, nout, nh);
    }
    // 2) fold attention vectors into weights
    egat_fold<<<1, 128, 0, stream>>>(W_src, b_src, W_dst, b_dst, W_edge, b_edge,
                                     att_src, att_dst, att_edge,
                                     w_as, w_ad, w_ae, csts);
    // 3) src feature GEMM on WMMA (fp32, exact)
    {
        int blocks = (N + 63) / 64;
        egat_gemm_src_wmma<<<blocks, 128, 0, stream>>>(x, W_src, b_src, src_feat, N);
    }
    // 4) per-node attention logits (folded 128->4 projections)
    {
        int blocks = ((N * 32) + 255) / 256;
        egat_node_alpha<<<blocks, 256, 0, stream>>>(x, w_as, w_ad, csts,
                                                    alpha_src, alpha_dst, N);
    }
    // 5) edge logits + segment max
    {
        int blocks = (E + 255) / 256;
        egat_edge_alpha<<<blocks, 256, 0, stream>>>(ei, edge_attr, alpha_src, alpha_dst,
                                                    w_ae, csts, alpha_e, m_max, E);
    }
    // 6) exp + segment sum
    {
        int blocks = (E + 255) / 256;
        egat_edge_exp<<<blocks, 256, 0, stream>>>(ei, m_max, alpha_e, s_sum, E);
    }
    // 7) weighted scatter of messages
    {
        long long total = (long long)E * 64;
        int blocks = (int)((total + 255) / 256);
        egat_scatter<<<blocks, 256, 0, stream>>>(ei, src_feat, alpha_e, s_sum, out, E);
    }
}